// DisplacementField_26414048870775
// MI455X (gfx1250) — compile-verified
//
#include <hip/hip_runtime.h>
#include <cstdint>
#include <cstddef>

#define FEAT    32
#define RES_H   512
#define RES_W   128
#define NCELL   (RES_H * RES_W)               // 65536 cells per plane
#define BINV    (-1.0f / 1.6f)                // -1/BOUNDS
#define TSTEP   (1.0f / (2.0f * (float)RES_W))
#define TPITCH  132                            // float4-aligned LDS pitch (128+4)

// ---- CDNA5 async global->LDS paths (guarded; graceful fallback) ----
#if defined(__gfx1250__) && __has_builtin(__builtin_amdgcn_global_load_async_to_lds_b32)
#define USE_ASYNC_LDS 1
typedef __attribute__((address_space(1))) int GAS;   // global (AS1) int*
typedef __attribute__((address_space(3))) int LAS;   // LDS (AS3) int*
#if __has_builtin(__builtin_amdgcn_global_load_async_to_lds_b128)
#define USE_ASYNC_LDS_B128 1
typedef int v4i_t __attribute__((vector_size(4 * sizeof(int))));
typedef __attribute__((address_space(1))) v4i_t GAS128;  // global (AS1) v4i*
typedef __attribute__((address_space(3))) v4i_t LAS128;  // LDS (AS3) v4i*
#else
#define USE_ASYNC_LDS_B128 0
#endif
#else
#define USE_ASYNC_LDS 0
#define USE_ASYNC_LDS_B128 0
#endif

__device__ __forceinline__ void wait_async_zero() {
#if USE_ASYNC_LDS
#if __has_builtin(__builtin_amdgcn_s_wait_asynccnt)
    __builtin_amdgcn_s_wait_asynccnt(0);
#else
    asm volatile("s_wait_asynccnt 0" ::: "memory");
#endif
#endif
}

// Nearest-neighbor index, align_corners=True, round-half-even (matches jnp.round)
__device__ __forceinline__ int coord_to_idx(float c, int size) {
    float v = rintf((c + 1.0f) * 0.5f * (float)(size - 1));
    v = fminf(fmaxf(v, 0.0f), (float)(size - 1));
    return (int)v;
}

// ---------------------------------------------------------------------------
// Kernel 1: transpose planes [F,H,W] -> ws[plane][h*W+w][F]
// One block per (plane, h). Load phase: 16 KB tile via async global->LDS
// (b128 per thread when available). Write phase: each thread gathers 4
// consecutive features from LDS and emits one 16B store; a wave covers
// 4 cells x 128B fully contiguous.
// ---------------------------------------------------------------------------
__global__ __launch_bounds__(1024)
void transpose_planes_kernel(const float* __restrict__ p0,
                             const float* __restrict__ p1,
                             const float* __restrict__ p2,
                             float* __restrict__ ws)
{
    __shared__ float tile[FEAT * TPITCH];
    int b     = blockIdx.x;
    int plane = b >> 9;            // /RES_H
    int h     = b & (RES_H - 1);
    const float* src = (plane == 0) ? p0 : ((plane == 1) ? p1 : p2);
    float* dst = ws + (size_t)plane * (size_t)NCELL * FEAT;
    int t = threadIdx.x;

#if USE_ASYNC_LDS_B128
    {   // one b128 async load per thread covers the whole 32x128 tile
        int f = t >> 5;                   // 0..31
        int w = (t & 31) * 4;             // 0..124 step 4 (lane dim -> coalesced)
        const float* g = src + (size_t)f * NCELL + (size_t)h * RES_W + w;
        float* l = &tile[f * TPITCH + w];
        __builtin_amdgcn_global_load_async_to_lds_b128((GAS128*)g, (LAS128*)l, 0, 0);
    }
#else
#pragma unroll
    for (int i = 0; i < 4; ++i) {
        int idx = t + i * 1024;           // 0..4095
        int f = idx >> 7;                 // 0..31
        int w = idx & (RES_W - 1);        // 0..127  (lane dim -> coalesced)
        const float* g = src + (size_t)f * NCELL + (size_t)h * RES_W + w;
        float* l = &tile[f * TPITCH + w];
#if USE_ASYNC_LDS
        __builtin_amdgcn_global_load_async_to_lds_b32((GAS*)g, (LAS*)l, 0, 0);
#else
        *l = *g;
#endif
    }
#endif
    wait_async_zero();
    __syncthreads();

    {   // write phase: q = feature-quad, w = cell column
        int q  = t & 7;                   // 0..7  -> features 4q..4q+3
        int w  = t >> 3;                  // 0..127
        int f0 = q * 4;
        float4 v;
        v.x = tile[(f0 + 0) * TPITCH + w];
        v.y = tile[(f0 + 1) * TPITCH + w];
        v.z = tile[(f0 + 2) * TPITCH + w];
        v.w = tile[(f0 + 3) * TPITCH + w];
        float4* d4 = reinterpret_cast<float4*>(dst + (size_t)(h * RES_W + w) * FEAT);
        d4[q] = v;
    }
}

// ---------------------------------------------------------------------------
// Kernel 2: main gather+product. 8 lanes per point, float4 per lane:
// each wave handles 4 points with b128 loads/stores (512B contiguous stores).
// ---------------------------------------------------------------------------
__global__ __launch_bounds__(256)
void sample_mul_kernel(const float* __restrict__ pts,
                       const float* __restrict__ tm,
                       const float* __restrict__ p0,
                       const float* __restrict__ p1,
                       const float* __restrict__ p2,
                       const float* __restrict__ wsP, int use_ws,
                       float* __restrict__ outA, float* __restrict__ outB,
                       int npts)
{
    int gid = blockIdx.x * blockDim.x + threadIdx.x;
    int n   = gid >> 3;                        // point index (8 lanes per point)
    int q   = gid & 7;                         // feature-quad index
    if (n >= npts) return;

    if (n + 1024 < npts) {                     // stream prefetch of upcoming points
        __builtin_prefetch(&pts[(size_t)3 * (n + 1024)], 0, 0);
    }

    float px = pts[3 * n + 0] * BINV;
    float py = pts[3 * n + 1] * BINV;
    float pz = pts[3 * n + 2] * BINV;
    float t  = tm[n] * 2.0f - 1.0f;

    int ix  = coord_to_idx(t,  RES_W);
    int c0  = coord_to_idx(px, RES_H) * RES_W + ix;
    int c1  = coord_to_idx(py, RES_H) * RES_W + ix;
    int c2  = coord_to_idx(pz, RES_H) * RES_W + ix;

    float4 v0, v1, v2;
    if (use_ws) {
        const float4* w4 = reinterpret_cast<const float4*>(wsP);
        const size_t PLANE4 = (size_t)NCELL * (FEAT / 4);     // float4s per plane
        v0 = w4[(size_t)c0 * (FEAT / 4) + q];
        v1 = w4[PLANE4     + (size_t)c1 * (FEAT / 4) + q];
        v2 = w4[PLANE4 * 2 + (size_t)c2 * (FEAT / 4) + q];
    } else {
        int f0 = q * 4;                        // strided fallback on [F,H,W]
        v0.x = p0[(size_t)(f0 + 0) * NCELL + c0];
        v0.y = p0[(size_t)(f0 + 1) * NCELL + c0];
        v0.z = p0[(size_t)(f0 + 2) * NCELL + c0];
        v0.w = p0[(size_t)(f0 + 3) * NCELL + c0];
        v1.x = p1[(size_t)(f0 + 0) * NCELL + c1];
        v1.y = p1[(size_t)(f0 + 1) * NCELL + c1];
        v1.z = p1[(size_t)(f0 + 2) * NCELL + c1];
        v1.w = p1[(size_t)(f0 + 3) * NCELL + c1];
        v2.x = p2[(size_t)(f0 + 0) * NCELL + c2];
        v2.y = p2[(size_t)(f0 + 1) * NCELL + c2];
        v2.z = p2[(size_t)(f0 + 2) * NCELL + c2];
        v2.w = p2[(size_t)(f0 + 3) * NCELL + c2];
    }

    float4 prod;
    prod.x = v0.x * v1.x * v2.x;
    prod.y = v0.y * v1.y * v2.y;
    prod.z = v0.z * v1.z * v2.z;
    prod.w = v0.w * v1.w * v2.w;

    size_t o4 = (size_t)n * (FEAT / 4) + q;
    reinterpret_cast<float4*>(outA)[o4] = prod;   // A == B except last row
    reinterpret_cast<float4*>(outB)[o4] = prod;   // (fixup rewrites that row)
}

// ---------------------------------------------------------------------------
// Kernel 3: fixup for the last point (data.at[-1].add(-TIME_STEP) quirk).
// ---------------------------------------------------------------------------
__device__ __forceinline__ float sample_product_scalar(
    float px, float py, float pz, float t, int f,
    const float* __restrict__ p0, const float* __restrict__ p1,
    const float* __restrict__ p2, const float* __restrict__ wsP, int use_ws)
{
    int ix  = coord_to_idx(t,  RES_W);
    int c0  = coord_to_idx(px, RES_H) * RES_W + ix;
    int c1  = coord_to_idx(py, RES_H) * RES_W + ix;
    int c2  = coord_to_idx(pz, RES_H) * RES_W + ix;
    float v0, v1, v2;
    if (use_ws) {
        v0 = wsP[(size_t)c0 * FEAT + f];
        v1 = wsP[(size_t)NCELL * FEAT     + (size_t)c1 * FEAT + f];
        v2 = wsP[(size_t)NCELL * FEAT * 2 + (size_t)c2 * FEAT + f];
    } else {
        v0 = p0[(size_t)f * NCELL + c0];
        v1 = p1[(size_t)f * NCELL + c1];
        v2 = p2[(size_t)f * NCELL + c2];
    }
    return v0 * v1 * v2;
}

__global__ __launch_bounds__(32)
void fixup_last_kernel(const float* __restrict__ pts,
                       const float* __restrict__ tm,
                       const float* __restrict__ p0,
                       const float* __restrict__ p1,
                       const float* __restrict__ p2,
                       const float* __restrict__ wsP, int use_ws,
                       float* __restrict__ outA, float* __restrict__ outB,
                       int npts)
{
    int f = threadIdx.x;
    int n = npts - 1;
    float px = pts[3 * n + 0] * BINV;
    float py = pts[3 * n + 1] * BINV;
    float pz = pts[3 * n + 2] * BINV;
    float t  = tm[n] * 2.0f - 1.0f;

    float a = sample_product_scalar(px, py, pz, t, f, p0, p1, p2, wsP, use_ws);
    float b = sample_product_scalar(px - TSTEP, py - TSTEP, pz - TSTEP, t - TSTEP,
                                    f, p0, p1, p2, wsP, use_ws);

    bool cond = (px + TSTEP) > 1.0f;           // scalar cond from last point's x
    float A = cond ? b : a;
    float B = cond ? a : b;
    size_t o = (size_t)n * FEAT + f;
    outA[o] = A;
    outB[o] = B;
}

extern "C" void kernel_launch(void* const* d_in, const int* in_sizes, int n_in,
                              void* d_out, int out_size, void* d_ws, size_t ws_size,
                              hipStream_t stream)
{
    const float* pts = (const float*)d_in[0];
    const float* tm  = (const float*)d_in[1];
    const float* p0  = (const float*)d_in[2];
    const float* p1  = (const float*)d_in[3];
    const float* p2  = (const float*)d_in[4];

    int npts = in_sizes[1];                    // time has N elements
    float* outA = (float*)d_out;
    float* outB = outA + (size_t)npts * FEAT;

    size_t need = (size_t)3 * NCELL * FEAT * sizeof(float);   // 24 MB
    int use_ws = (d_ws != nullptr && ws_size >= need) ? 1 : 0;
    float* wsP = (float*)d_ws;

    if (use_ws) {
        transpose_planes_kernel<<<3 * RES_H, 1024, 0, stream>>>(p0, p1, p2, wsP);
    }

    long long total = (long long)npts * (FEAT / 4);           // 8 lanes per point
    int blocks = (int)((total + 255) / 256);
    sample_mul_kernel<<<blocks, 256, 0, stream>>>(pts, tm, p0, p1, p2, wsP, use_ws,
                                                  outA, outB, npts);
    fixup_last_kernel<<<1, 32, 0, stream>>>(pts, tm, p0, p1, p2, wsP, use_ws,
                                            outA, outB, npts);
}